// MagneticRealAgnosticSpinOrbitCoupledDensityInteractionBlock_42125039239750
// MI455X (gfx1250) — compile-verified
//
#include <hip/hip_runtime.h>
#include <hip/hip_bf16.h>
#include <math.h>

typedef __attribute__((ext_vector_type(2))) float v2f;
typedef __attribute__((ext_vector_type(8))) float v8f;

#if __has_builtin(__builtin_amdgcn_global_load_async_to_lds_b128) && \
    __has_builtin(__builtin_amdgcn_s_wait_asynccnt)
#define USE_ASYNC_LDS 1
typedef __attribute__((vector_size(16))) int vi4;
typedef __attribute__((address_space(1))) vi4 gvi4;
typedef __attribute__((address_space(3))) vi4 lvi4;
#else
#define USE_ASYNC_LDS 0
#endif

__device__ __forceinline__ float silu_fast(float x)
{
    // x * sigmoid(x); v_exp_f32 + v_rcp_f32 instead of IEEE divide expansion
    return x * __builtin_amdgcn_rcpf(1.0f + __expf(-x));
}

// ---------------------------------------------------------------------------
// fp32 WMMA GEMM: C[r*ldc + c*ces] = act( scale * sum_k A[r*lda+k*aes]*B[k*ldb+c] )
// One wave computes a 16 x (16*TNB) output tile via V_WMMA_F32_16X16X4_F32.
// A-frag (16x4 f32): lanes 0-15 row M=lane, v0=K0,v1=K1; lanes16-31 v0=K2,v1=K3.
// B-frag (4x16):     v0 lanes0-15 = row K0, lanes16-31 = row K2; v1 = K1/K3.
// C/D 16x16: vgpr r, lanes0-15 -> (M=r, N=lane); lanes16-31 -> (M=r+8, N=lane-16).
// A fragment is loaded once per K-step and reused across TNB B subtiles.
// ---------------------------------------------------------------------------
template <int TNB, bool STRIDED>
__global__ void gemm_f32_wmma(const float* __restrict__ A, int lda, int aes,
                              const float* __restrict__ B, int ldb,
                              float* __restrict__ C, int ldc, int ces,
                              int M, int K, int Nc, float scale, int act)
{
    int wave   = (int)((blockIdx.x * blockDim.x + threadIdx.x) >> 5);
    int lane   = threadIdx.x & 31;
    int tilesN = Nc / (16 * TNB);
    int tm     = wave / tilesN;
    int tn     = wave - tm * tilesN;
    if (tm * 16 >= M) return;                 // uniform per wave: EXEC stays all-ones
    int hi = lane >> 4;                       // 0 or 1
    int lo = lane & 15;

    const float* Arow = A + (size_t)(tm * 16 + lo) * lda;
    const float* Bcol = B + (size_t)(tn * 16 * TNB + lo);

    // warm caches (global_prefetch_b8)
    __builtin_prefetch(Arow, 0, 1);
    __builtin_prefetch(Bcol, 0, 1);

    v8f acc[TNB];
#pragma unroll
    for (int t = 0; t < TNB; ++t) acc[t] = (v8f){0.f,0.f,0.f,0.f,0.f,0.f,0.f,0.f};

#pragma unroll 2
    for (int k0 = 0; k0 < K; k0 += 4) {
        int ka = k0 + 2 * hi;
        v2f a;
        if (STRIDED) {
            a.x = Arow[(size_t)ka * aes];
            a.y = Arow[(size_t)(ka + 1) * aes];
        } else {
            a = *(const v2f*)(Arow + ka);     // contiguous pair -> one b64 load
        }
        const float* B0 = Bcol + (size_t)ka * ldb;
        const float* B1 = Bcol + (size_t)(ka + 1) * ldb;
#pragma unroll
        for (int t = 0; t < TNB; ++t) {
            v2f b;
            b.x = B0[t * 16];
            b.y = B1[t * 16];
            acc[t] = __builtin_amdgcn_wmma_f32_16x16x4_f32(false, a, false, b,
                                                           (short)0, acc[t], false, false);
        }
    }
#pragma unroll
    for (int t = 0; t < TNB; ++t) {
#pragma unroll
        for (int r = 0; r < 8; ++r) {
            int row = tm * 16 + r + 8 * hi;
            int col = tn * 16 * TNB + t * 16 + lo;
            float x = acc[t][r] * scale;
            if (act) x = silu_fast(x);
            C[(size_t)row * ldc + (size_t)col * ces] = x;
        }
    }
}

// ---------------------------------------------------------------------------
// Edge prep: efm = [edge_feats | magmom_node_inv_feats[sender]],
// density[receiver] += tanh((edge_feats . dens_w / sqrt(8))^2)
// ---------------------------------------------------------------------------
__global__ void edge_prep(const float* __restrict__ edge_feats,
                          const float* __restrict__ mm_inv,
                          const int*   __restrict__ edge_index,
                          const float* __restrict__ dens_w,
                          float* __restrict__ efm,
                          float* __restrict__ density, int E)
{
    int e = blockIdx.x * blockDim.x + threadIdx.x;
    if (e >= E) return;
    int snd = edge_index[e];
    int rcv = edge_index[E + e];
    float d = 0.f;
#pragma unroll
    for (int j = 0; j < 8; ++j) {
        float ef = edge_feats[(size_t)e * 8 + j];
        efm[(size_t)e * 16 + j]     = ef;
        efm[(size_t)e * 16 + 8 + j] = mm_inv[(size_t)snd * 8 + j];
        d += ef * dens_w[j];
    }
    d *= 0.35355339059327373f; // 1/sqrt(8)
    d = tanhf(d * d);
    atomicAdd(&density[rcv], d);
}

// ---------------------------------------------------------------------------
// Fused per-edge tensor products + magnetic weighting + scatter to node accs.
// One thread = (edge e, channel u). mag_w (16x832) staged in LDS (52 KB),
// filled via the CDNA5 async global->LDS DMA path when available.
// ---------------------------------------------------------------------------
__global__ void edge_tensor_scatter(const float* __restrict__ s_up,
                                    const float* __restrict__ v_up,   // 3 planes N*64
                                    const float* __restrict__ tp_w,   // E*320
                                    const float* __restrict__ efm,    // E*16
                                    const float* __restrict__ edge_attrs,
                                    const float* __restrict__ mag_attrs,
                                    const int*   __restrict__ edge_index,
                                    const float* __restrict__ mag_w,  // 16*832
                                    float* __restrict__ ms_acc,       // N*320
                                    float* __restrict__ mv_acc,       // N*1536
                                    int E, int Nn)
{
    __shared__ float smw[16 * 832];
#if USE_ASYNC_LDS
    // bulk LDS fill without touching VGPRs: GLOBAL_LOAD_ASYNC_TO_LDS_B128
    for (int i = threadIdx.x * 4; i < 16 * 832; i += blockDim.x * 4) {
        __builtin_amdgcn_global_load_async_to_lds_b128(
            (gvi4*)(mag_w + i), (lvi4*)(smw + i), 0, 0);
    }
    __builtin_amdgcn_s_wait_asynccnt(0);
    __syncthreads();
#else
    for (int i = threadIdx.x; i < 16 * 832; i += blockDim.x) smw[i] = mag_w[i];
    __syncthreads();
#endif

    int t = blockIdx.x * blockDim.x + threadIdx.x;
    int e = t >> 6;
    int u = t & 63;
    if (e >= E) return;

    int snd = edge_index[e];
    int rcv = edge_index[E + e];

    float xs = s_up[(size_t)snd * 64 + u];
    float xv[3];
#pragma unroll
    for (int i = 0; i < 3; ++i) xv[i] = v_up[(size_t)i * Nn * 64 + (size_t)snd * 64 + u];

    float y0 = edge_attrs[(size_t)e * 4 + 0];
    float y1[3] = { edge_attrs[(size_t)e * 4 + 1],
                    edge_attrs[(size_t)e * 4 + 2],
                    edge_attrs[(size_t)e * 4 + 3] };

    float tp[5];
#pragma unroll
    for (int p = 0; p < 5; ++p) tp[p] = tp_w[(size_t)e * 320 + p * 64 + u];

    const float INV_SQ3 = 0.5773502691896258f;
    const float INV_SQ2 = 0.7071067811865476f;

    // conventional tensor products
    float m_s0 = tp[0] * xs * y0;
    float m_v0[3], m_v1[3], m_v2[3];
#pragma unroll
    for (int i = 0; i < 3; ++i) {
        m_v0[i] = tp[1] * xs * y1[i];
        m_v1[i] = tp[2] * xv[i] * y0;
    }
    float dotv = xv[0] * y1[0] + xv[1] * y1[1] + xv[2] * y1[2];
    float m_s1 = tp[3] * dotv * INV_SQ3;
    m_v2[0] = tp[4] * (xv[1] * y1[2] - xv[2] * y1[1]) * INV_SQ2;
    m_v2[1] = tp[4] * (xv[2] * y1[0] - xv[0] * y1[2]) * INV_SQ2;
    m_v2[2] = tp[4] * (xv[0] * y1[1] - xv[1] * y1[0]) * INV_SQ2;

    // magnetic weights wm[13] = (efm . mag_w[:,p*64+u]) / sqrt(16)
    float fe[16];
#pragma unroll
    for (int k = 0; k < 16; ++k) fe[k] = efm[(size_t)e * 16 + k];
    float wm[13];
#pragma unroll
    for (int p = 0; p < 13; ++p) {
        float s = 0.f;
#pragma unroll
        for (int k = 0; k < 16; ++k) s += fe[k] * smw[k * 832 + p * 64 + u];
        wm[p] = s * 0.25f;
    }

    float m0 = mag_attrs[(size_t)snd * 4 + 0];
    float m1[3] = { mag_attrs[(size_t)snd * 4 + 1],
                    mag_attrs[(size_t)snd * 4 + 2],
                    mag_attrs[(size_t)snd * 4 + 3] };

    float out_s[5];
    float out_v[8][3];

    // scalar-path terms
    out_s[0] = wm[0] * m_s0 * m0;
    out_s[1] = wm[2] * m_s1 * m0;
#pragma unroll
    for (int i = 0; i < 3; ++i) {
        out_v[0][i] = wm[1] * m_s0 * m1[i];
        out_v[1][i] = wm[3] * m_s1 * m1[i];
    }
    // vector-path terms: vv in (m_v0, m_v1, m_v2), p = 4,7,10
    const float* vvs[3] = { m_v0, m_v1, m_v2 };
#pragma unroll
    for (int j = 0; j < 3; ++j) {
        const float* vv = vvs[j];
        int p = 4 + 3 * j;
#pragma unroll
        for (int i = 0; i < 3; ++i) out_v[2 + 2 * j][i] = wm[p] * vv[i] * m0;
        float dvm = vv[0] * m1[0] + vv[1] * m1[1] + vv[2] * m1[2];
        out_s[2 + j] = wm[p + 1] * dvm * INV_SQ3;
        out_v[3 + 2 * j][0] = wm[p + 2] * (vv[1] * m1[2] - vv[2] * m1[1]) * INV_SQ2;
        out_v[3 + 2 * j][1] = wm[p + 2] * (vv[2] * m1[0] - vv[0] * m1[2]) * INV_SQ2;
        out_v[3 + 2 * j][2] = wm[p + 2] * (vv[0] * m1[1] - vv[1] * m1[0]) * INV_SQ2;
    }

    // scatter (segment sum over receiver)
    float* msr = ms_acc + (size_t)rcv * 320 + u;
    float* mvr = mv_acc + (size_t)rcv * 1536 + (size_t)u * 3;
#pragma unroll
    for (int q = 0; q < 5; ++q) atomicAdd(msr + q * 64, out_s[q]);
#pragma unroll
    for (int q = 0; q < 8; ++q)
#pragma unroll
        for (int i = 0; i < 3; ++i) atomicAdd(mvr + q * 192 + i, out_v[q][i]);
}

// ---------------------------------------------------------------------------
// Build outer-product inputs for the skip contraction, folding in the
// density normalization: U[n, u*10+v] = (ms[n,u]/(density+1)) * node_attrs[n,v]
// 4 planes of N*640: plane0 = scalar, planes 1..3 = vector components.
// ---------------------------------------------------------------------------
__global__ void build_skip_inputs(const float* __restrict__ ms_lin,  // N*64
                                  const float* __restrict__ mv_lin,  // 3 planes N*64
                                  const float* __restrict__ density, // N
                                  const float* __restrict__ node_attrs, // N*10
                                  float* __restrict__ U, int Nn)
{
    int t = blockIdx.x * blockDim.x + threadIdx.x;
    if (t >= Nn * 640) return;
    int n = t / 640;
    int j = t - n * 640;
    int u = j / 10;
    int v = j - u * 10;
    float inv = __builtin_amdgcn_rcpf(density[n] + 1.0f);
    float av  = node_attrs[(size_t)n * 10 + v];
    U[t] = ms_lin[(size_t)n * 64 + u] * inv * av;
#pragma unroll
    for (int i = 0; i < 3; ++i)
        U[(size_t)(i + 1) * Nn * 640 + t] =
            mv_lin[(size_t)i * Nn * 64 + (size_t)n * 64 + u] * inv * av;
}

// ---------------------------------------------------------------------------
static inline void launch_gemm(const float* A, int lda, int aes,
                               const float* B, int ldb,
                               float* C, int ldc, int ces,
                               int M, int K, int Nc, float scale, int act,
                               hipStream_t stream)
{
    bool strided = (aes != 1);
    if ((Nc & 63) == 0) {
        long tiles  = (long)(M / 16) * (Nc / 64);
        int  blocks = (int)((tiles + 7) / 8);     // 8 waves (256 threads) per block
        if (strided)
            gemm_f32_wmma<4, true><<<blocks, 256, 0, stream>>>(A, lda, aes, B, ldb,
                                                               C, ldc, ces, M, K, Nc, scale, act);
        else
            gemm_f32_wmma<4, false><<<blocks, 256, 0, stream>>>(A, lda, aes, B, ldb,
                                                                C, ldc, ces, M, K, Nc, scale, act);
    } else {
        long tiles  = (long)(M / 16) * (Nc / 16);
        int  blocks = (int)((tiles + 7) / 8);
        if (strided)
            gemm_f32_wmma<1, true><<<blocks, 256, 0, stream>>>(A, lda, aes, B, ldb,
                                                               C, ldc, ces, M, K, Nc, scale, act);
        else
            gemm_f32_wmma<1, false><<<blocks, 256, 0, stream>>>(A, lda, aes, B, ldb,
                                                                C, ldc, ces, M, K, Nc, scale, act);
    }
}

extern "C" void kernel_launch(void* const* d_in, const int* in_sizes, int n_in,
                              void* d_out, int out_size, void* d_ws, size_t ws_size,
                              hipStream_t stream)
{
    const float* node_attrs = (const float*)d_in[0];
    const float* node_feats = (const float*)d_in[1];
    const float* edge_attrs = (const float*)d_in[2];
    const float* edge_feats = (const float*)d_in[3];
    const int*   edge_index = (const int*)d_in[4];
    const float* mm_inv     = (const float*)d_in[5];
    const float* mag_attrs  = (const float*)d_in[6];
    const float* W_up_s     = (const float*)d_in[7];
    const float* W_up_v     = (const float*)d_in[8];
    const float* rmlp_w0    = (const float*)d_in[9];
    const float* rmlp_w1    = (const float*)d_in[10];
    const float* rmlp_w2    = (const float*)d_in[11];
    const float* rmlp_w3    = (const float*)d_in[12];
    const float* mag_w      = (const float*)d_in[13];
    const float* dens_w     = (const float*)d_in[14];
    const float* lin_s_w    = (const float*)d_in[15];
    const float* lin_v_w    = (const float*)d_in[16];
    const float* skip_w_s   = (const float*)d_in[17];
    const float* skip_w_v   = (const float*)d_in[18];
    float*       out        = (float*)d_out;

    const int Nn = in_sizes[1] / 256;   // node_feats: N x (4*MUL)
    const int E  = in_sizes[3] / 8;     // edge_feats: E x NR

    // ---- workspace layout (floats) ----
    float* w      = (float*)d_ws;
    float* s_up   = w;                            // N*64
    float* v_up   = s_up  + (size_t)Nn * 64;      // 3*N*64
    float* efm    = v_up  + (size_t)3 * Nn * 64;  // E*16
    float* h_a    = efm   + (size_t)E * 16;       // E*64
    float* h_b    = h_a   + (size_t)E * 64;       // E*64
    float* tp     = h_b   + (size_t)E * 64;       // max(E*320, 4*N*640); reused as U
    size_t tpsz   = (size_t)E * 320;
    size_t usz    = (size_t)4 * Nn * 640;
    size_t big    = tpsz > usz ? tpsz : usz;
    float* dens   = tp    + big;                  // N
    float* ms_acc = dens  + (size_t)Nn;           // N*320
    float* mv_acc = ms_acc + (size_t)Nn * 320;    // N*1536
    float* ms_lin = mv_acc + (size_t)Nn * 1536;   // N*64
    float* mv_lin = ms_lin + (size_t)Nn * 64;     // 3*N*64
    float* U      = tp;

    const float inv_s64  = 0.125f;                 // 1/sqrt(64)
    const float inv_s16  = 0.25f;                  // 1/sqrt(16)
    const float inv_s320 = 0.05590169943749474f;   // 1/sqrt(320)
    const float inv_s512 = 0.04419417382415922f;   // 1/sqrt(512)
    const float inv_fan  = 0.03952847075210474f;   // 1/sqrt(640)

    // 0) zero segment-sum accumulators (dens..mv_acc are contiguous)
    (void)hipMemsetAsync(dens, 0, (size_t)(Nn + Nn * 320 + Nn * 1536) * sizeof(float), stream);

    // 1) node up-projection (WMMA)
    launch_gemm(node_feats, 256, 1, W_up_s, 64, s_up, 64, 1, Nn, 64, 64, inv_s64, 0, stream);
    for (int i = 0; i < 3; ++i)
        launch_gemm(node_feats + 64 + i, 256, 3, W_up_v, 64,
                    v_up + (size_t)i * Nn * 64, 64, 1, Nn, 64, 64, inv_s64, 0, stream);

    // 2) edge features + density scatter
    edge_prep<<<(E + 255) / 256, 256, 0, stream>>>(edge_feats, mm_inv, edge_index,
                                                   dens_w, efm, dens, E);

    // 3) radial MLP (WMMA + SiLU)
    launch_gemm(efm, 16, 1, rmlp_w0, 64, h_a, 64, 1, E, 16, 64, inv_s16, 1, stream);
    launch_gemm(h_a, 64, 1, rmlp_w1, 64, h_b, 64, 1, E, 64, 64, inv_s64, 1, stream);
    launch_gemm(h_b, 64, 1, rmlp_w2, 64, h_a, 64, 1, E, 64, 64, inv_s64, 1, stream);
    launch_gemm(h_a, 64, 1, rmlp_w3, 320, tp, 320, 1, E, 64, 320, inv_s64, 0, stream);

    // 4) fused per-edge tensor products + magnetic weighting + scatter
    edge_tensor_scatter<<<(E * 64 + 255) / 256, 256, 0, stream>>>(
        s_up, v_up, tp, efm, edge_attrs, mag_attrs, edge_index, mag_w,
        ms_acc, mv_acc, E, Nn);

    // 5) post-aggregation linears (WMMA)
    launch_gemm(ms_acc, 320, 1, lin_s_w, 64, ms_lin, 64, 1, Nn, 320, 64, inv_s320, 0, stream);
    for (int i = 0; i < 3; ++i)
        launch_gemm(mv_acc + i, 1536, 3, lin_v_w, 64,
                    mv_lin + (size_t)i * Nn * 64, 64, 1, Nn, 512, 64, inv_s512, 0, stream);

    // 6) density normalization + outer products for skip contraction
    build_skip_inputs<<<(Nn * 640 + 255) / 256, 256, 0, stream>>>(
        ms_lin, mv_lin, dens, node_attrs, U, Nn);

    // 7) skip contraction (WMMA, K=640) writing directly into (N,64,4) output
    launch_gemm(U, 640, 1, skip_w_s, 64, out + 0, 256, 4, Nn, 640, 64, inv_fan, 0, stream);
    for (int i = 0; i < 3; ++i)
        launch_gemm(U + (size_t)(i + 1) * Nn * 640, 640, 1, skip_w_v, 64,
                    out + 1 + i, 256, 4, Nn, 640, 64, inv_fan, 0, stream);
}